// GAT_27084063768797
// MI455X (gfx1250) — compile-verified
//
#include <hip/hip_runtime.h>

typedef float v2f __attribute__((ext_vector_type(2)));
typedef float v8f __attribute__((ext_vector_type(8)));

#define NEG_SLOPE 0.2f
#define EPS_DENOM 1e-16f

// ---------------- ordered-uint encoding for float atomic max ----------------
__device__ __forceinline__ unsigned enc_f32(float f) {
    unsigned u = __float_as_uint(f);
    return ((int)u >= 0) ? (u | 0x80000000u) : ~u;
}
__device__ __forceinline__ float dec_f32(unsigned u) {
    return __uint_as_float(((int)u < 0) ? (u & 0x7FFFFFFFu) : ~u);
}

// ---------------- zero fill ----------------
__global__ void gat_zero_u32(unsigned* __restrict__ p, long long n) {
    long long i = blockIdx.x * (long long)blockDim.x + threadIdx.x;
    long long stride = (long long)gridDim.x * blockDim.x;
    for (; i < n; i += stride) p[i] = 0u;
}

// ---------------- W[K x M] -> Wt[M x K] (tiny, one-off per layer) ----------------
__global__ void gat_transpose(const float* __restrict__ W, float* __restrict__ Wt,
                              int K, int M) {
    int i = blockIdx.x * blockDim.x + threadIdx.x;
    if (i >= K * M) return;
    int k = i / M;
    int m = i - k * M;
    Wt[(size_t)m * K + k] = W[i];
}

// ---------------- fp32 WMMA GEMM: H[nrows x M] = act(A[nrows x K]) @ W[K x M] ----------------
// One wave computes a 16x16 tile via V_WMMA_F32_16X16X4_F32, K stepped by 4.
// B operand read from pre-transposed Wt[M x K]: one aligned b64 load per lane per step.
// Out-of-range columns: address clamped (loads always legal); since D[:,n] depends only
// on B[:,n] (per-lane), garbage columns are simply not stored — no zeroing needed.
template <bool RELUA>
__global__ void gat_gemm_wmma(const float* __restrict__ A, const float* __restrict__ Wt,
                              float* __restrict__ H, int nrows, int K, int M,
                              int colTiles) {
    int wave = threadIdx.x >> 5;
    int lane = threadIdx.x & 31;
    int rowTiles = nrows >> 4;                  // nrows is a multiple of 16
    int tile = blockIdx.x * (blockDim.x >> 5) + wave;
    if (tile >= rowTiles * colTiles) return;

    int rt = tile / colTiles;
    int ct = tile - rt * colTiles;
    int row0 = rt << 4;
    int col0 = ct << 4;

    int half = lane >> 4;        // 0: K-pair {k,k+1}, 1: K-pair {k+2,k+3}
    int l15  = lane & 15;        // A row within tile / B,C column within tile
    int koff = half << 1;

    const float* __restrict__ arow = A + (size_t)(row0 + l15) * K + koff;
    int nIdx   = col0 + l15;
    int nClamp = (nIdx < M) ? nIdx : (M - 1);
    const float* __restrict__ wrow = Wt + (size_t)nClamp * K + koff;

    v8f acc = {};
#pragma unroll 4
    for (int k = 0; k < K; k += 4) {
        v2f a = *(const v2f*)(arow + k);        // 8B aligned: K%4==0, koff even
        if (RELUA) { a.x = fmaxf(a.x, 0.0f); a.y = fmaxf(a.y, 0.0f); }
        v2f b = *(const v2f*)(wrow + k);
        acc = __builtin_amdgcn_wmma_f32_16x16x4_f32(
            /*neg_a=*/false, a, /*neg_b=*/false, b,
            /*c_mod=*/(short)0, acc, /*reuse_a=*/false, /*reuse_b=*/false);
    }
    if (nIdx < M) {
        int rbase = row0 + (half << 3);   // lanes 0-15 -> rows 0..7, lanes 16-31 -> 8..15
#pragma unroll
        for (int r = 0; r < 8; ++r)
            H[(size_t)(rbase + r) * M + nIdx] = acc[r];
    }
}

// ---------------- per-node attention coefficients: asrc[i]=h_i.a_src, adst[i]=h_i.a_dst ----------------
__global__ void gat_attn(const float* __restrict__ H, const float* __restrict__ avs,
                         const float* __restrict__ avd, float* __restrict__ asrc,
                         float* __restrict__ adst, int N, int M) {
    int node = (int)((blockIdx.x * (long long)blockDim.x + threadIdx.x) >> 5);
    int lane = threadIdx.x & 31;
    if (node >= N) return;
    const float* __restrict__ h = H + (size_t)node * M;
    float s = 0.0f, d = 0.0f;
    for (int f = lane; f < M; f += 32) {
        float hv = h[f];
        s += hv * avs[f];
        d += hv * avd[f];
    }
#pragma unroll
    for (int off = 16; off; off >>= 1) {
        s += __shfl_down(s, off, 32);
        d += __shfl_down(d, off, 32);
    }
    if (lane == 0) { asrc[node] = s; adst[node] = d; }
}

// ---------------- edge pass 1: e = leaky_relu(asrc[s]+adst[d]); segment max via atomic u32 max ----------------
__global__ void gat_edge_max(const long long* __restrict__ ei, long long E, long long ET,
                             const float* __restrict__ asrc, const float* __restrict__ adst,
                             float* __restrict__ ebuf, unsigned* __restrict__ mmax) {
    long long t = blockIdx.x * (long long)blockDim.x + threadIdx.x;
    if (t >= ET) return;
    int s, d;
    if (t < E) { s = (int)ei[t]; d = (int)ei[E + t]; }
    else       { s = d = (int)(t - E); }          // appended self loops
    float e = asrc[s] + adst[d];
    e = (e > 0.0f) ? e : NEG_SLOPE * e;
    ebuf[t] = e;
    atomicMax(mmax + d, enc_f32(e));
}

// ---------------- edge pass 2: ex = exp(e - m[d]); segment sum ----------------
__global__ void gat_edge_exp(const long long* __restrict__ ei, long long E, long long ET,
                             const unsigned* __restrict__ mmax, float* __restrict__ ebuf,
                             float* __restrict__ denom) {
    long long t = blockIdx.x * (long long)blockDim.x + threadIdx.x;
    if (t >= ET) return;
    int d = (t < E) ? (int)ei[E + t] : (int)(t - E);
    float m = dec_f32(mmax[d]);
    float ex = __expf(ebuf[t] - m);
    ebuf[t] = ex;
    atomicAdd(denom + d, ex);
}

// ---------------- edge pass 3: out[d,:] += alpha * h[s,:] (edge x feature parallel) ----------------
// MT>0: compile-time feature width (shift/and or small const div); MT==0: generic.
template <int MT>
__global__ void gat_scatter(const long long* __restrict__ ei, long long E, long long ET,
                            const float* __restrict__ H, const float* __restrict__ ebuf,
                            const float* __restrict__ denom, float* __restrict__ out,
                            int Mrt) {
    const int M   = (MT > 0) ? MT : Mrt;
    const int epb = 256 / M;                       // edges per 256-thread block
    int el = threadIdx.x / M;                      // compile-time strength-reduced for MT>0
    int f  = threadIdx.x - el * M;
    if (el >= epb) return;
    long long t = blockIdx.x * (long long)epb + el;
    if (t >= ET) return;
    int s, d;
    if (t < E) { s = (int)ei[t]; d = (int)ei[E + t]; }
    else       { s = d = (int)(t - E); }
    float alpha = ebuf[t] / (denom[d] + EPS_DENOM);
    atomicAdd(out + (size_t)d * M + f, H[(size_t)s * M + f] * alpha);
}

// ---------------- orchestration ----------------
static inline size_t align256(size_t x) { return (x + 255) & ~(size_t)255; }

extern "C" void kernel_launch(void* const* d_in, const int* in_sizes, int n_in,
                              void* d_out, int out_size, void* d_ws, size_t ws_size,
                              hipStream_t stream) {
    (void)n_in; (void)ws_size; (void)out_size;
    const float*     x    = (const float*)d_in[0];
    const long long* ei   = (const long long*)d_in[1];   // int64 edge_index [2,E]
    const float*     W0   = (const float*)d_in[2];
    const float*     as0  = (const float*)d_in[3];
    const float*     ad0  = (const float*)d_in[4];
    const float*     W1   = (const float*)d_in[5];
    const float*     as1  = (const float*)d_in[6];
    const float*     ad1  = (const float*)d_in[7];
    const float*     W2   = (const float*)d_in[8];
    const float*     as2  = (const float*)d_in[9];
    const float*     ad2  = (const float*)d_in[10];

    const int Hdim = in_sizes[3];                 // 64
    const int Fin  = in_sizes[2] / Hdim;          // 256
    const int N    = in_sizes[0] / Fin;           // 100000
    const int C    = in_sizes[9];                 // 40
    const long long E  = in_sizes[1] / 2;         // 1.6M
    const long long ET = E + N;                   // + self loops

    // workspace carve-up
    char* ws = (char*)d_ws;
    size_t off = 0;
    float*    hbuf  = (float*)(ws + off); off += align256((size_t)N * Hdim * sizeof(float));
    float*    agg   = (float*)(ws + off); off += align256((size_t)N * Hdim * sizeof(float));
    float*    wtb   = (float*)(ws + off); off += align256((size_t)Fin * Hdim * sizeof(float));
    float*    asrcv = (float*)(ws + off); off += align256((size_t)N * sizeof(float));
    float*    adstv = (float*)(ws + off); off += align256((size_t)N * sizeof(float));
    unsigned* mmax  = (unsigned*)(ws + off); off += align256((size_t)N * sizeof(unsigned));
    float*    denom = (float*)(ws + off); off += align256((size_t)N * sizeof(float));
    float*    ebuf  = (float*)(ws + off); off += align256((size_t)ET * sizeof(float));

    const int TB = 256;
    auto blocks1d = [](long long n, int tb) { return (int)((n + tb - 1) / tb); };

    struct Layer { const float* A; int K; int M; int reluA; const float* W;
                   const float* avs; const float* avd; float* out; };
    Layer layers[3] = {
        { x,    Fin,  Hdim, 0, W0, as0, ad0, agg },
        { agg,  Hdim, Hdim, 1, W1, as1, ad1, agg },
        { agg,  Hdim, C,    1, W2, as2, ad2, (float*)d_out },
    };

    for (int l = 0; l < 3; ++l) {
        const Layer& L = layers[l];
        const int M = L.M;
        const int colTiles = (M + 15) / 16;
        const int rowTiles = N / 16;

        // 0) transpose W for b64 B-fragment loads
        gat_transpose<<<blocks1d((long long)L.K * M, TB), TB, 0, stream>>>(L.W, wtb, L.K, M);

        // 1) h = act(A) @ W   (fp32 WMMA, 8 waves/block, one 16x16 tile per wave)
        int tiles = rowTiles * colTiles;
        int gblk  = (tiles + 7) / 8;
        if (L.reluA)
            gat_gemm_wmma<true ><<<gblk, TB, 0, stream>>>(L.A, wtb, hbuf, N, L.K, M, colTiles);
        else
            gat_gemm_wmma<false><<<gblk, TB, 0, stream>>>(L.A, wtb, hbuf, N, L.K, M, colTiles);

        // 2) attention coefficients
        gat_attn<<<blocks1d((long long)N * 32, TB), TB, 0, stream>>>(hbuf, L.avs, L.avd, asrcv, adstv, N, M);

        // 3) clear segment-max / denom / output accumulator
        gat_zero_u32<<<blocks1d(N, TB), TB, 0, stream>>>(mmax, N);
        gat_zero_u32<<<blocks1d(N, TB), TB, 0, stream>>>((unsigned*)denom, N);
        gat_zero_u32<<<blocks1d((long long)N * M, TB), TB, 0, stream>>>((unsigned*)L.out, (long long)N * M);

        // 4) segment softmax + weighted scatter
        gat_edge_max<<<blocks1d(ET, TB), TB, 0, stream>>>(ei, E, ET, asrcv, adstv, ebuf, mmax);
        gat_edge_exp<<<blocks1d(ET, TB), TB, 0, stream>>>(ei, E, ET, mmax, ebuf, denom);

        int epb = 256 / M;
        int sblk = (int)((ET + epb - 1) / epb);
        if (M == 64)
            gat_scatter<64><<<sblk, TB, 0, stream>>>(ei, E, ET, hbuf, ebuf, denom, L.out, M);
        else if (M == 40)
            gat_scatter<40><<<sblk, TB, 0, stream>>>(ei, E, ET, hbuf, ebuf, denom, L.out, M);
        else
            gat_scatter<0><<<sblk, TB, 0, stream>>>(ei, E, ET, hbuf, ebuf, denom, L.out, M);
    }
}